// MS_Decoder_42606075576369
// MI455X (gfx1250) — compile-verified
//
#include <hip/hip_runtime.h>
#include <math.h>
#include <stdint.h>

#define N_ 1024
#define W_ 6
#define E_ (N_ * W_)

// ---------------------------------------------------------------------------
// All persistent state uses [index][B] layout so that lane = batch element:
// every gather index is uniform across a wave32 -> contiguous wave loads.
// ---------------------------------------------------------------------------

// channel (B x N) -> channelT (N x B), classic 32x33 LDS tile transpose.
__global__ void k_transpose(const float* __restrict__ src,
                            float* __restrict__ dst, int B) {
  __shared__ float tile[32][33];
  const int bx = blockIdx.x * 32;  // N tile
  const int by = blockIdx.y * 32;  // B tile
  const int tx = threadIdx.x, ty = threadIdx.y;
#pragma unroll
  for (int k = 0; k < 4; ++k) {
    const int b = by + ty + 8 * k;
    tile[ty + 8 * k][tx] = src[(size_t)b * N_ + (bx + tx)];
  }
  __syncthreads();
#pragma unroll
  for (int k = 0; k < 4; ++k) {
    const int v = bx + ty + 8 * k;
    dst[(size_t)v * B + (by + tx)] = tile[tx][ty + 8 * k];
  }
}

// llr_e init for iteration 0: e2o == 0 so llr_e[e] = channel[var(e)].
// float4: one 512B contiguous transaction per wave per row.
__global__ void k_init(const float* __restrict__ channelT,
                       const int* __restrict__ oddToEven,
                       float* __restrict__ llr_e, int B) {
  const int b4 = blockIdx.y * 128 + threadIdx.x * 4;
  const int e = blockIdx.x * 8 + threadIdx.y;
  const int v = oddToEven[e];
  *(float4*)(llr_e + (size_t)e * B + b4) =
      *(const float4*)(channelT + (size_t)v * B + b4);
}

// Check-node phase: e2o[e] = prod(sign(llr_e[peers])) * alpha * min|llr_e[peers]|
// The 5 peer-row gathers are staged LDS-ward through the CDNA5 async path
// (global_load_async_to_lds_b128, ASYNCcnt, s_wait_asynccnt) and consumed
// from LDS. Each wave stages/consumes only its own region -> no barrier.
__global__ void k_chk(const float* __restrict__ llr_e,
                      const int* __restrict__ edgeToChk,
                      const float* __restrict__ alpha,
                      float* __restrict__ e2o, int B) {
  __shared__ float4 stage[8][W_ - 1][32];  // 20 KB: [wave][peer][lane]
  const int tx = threadIdx.x, ty = threadIdx.y;
  const int b4 = blockIdx.y * 128 + tx * 4;   // 4 batch elems per lane
  const int e = blockIdx.x * 8 + ty;          // uniform per wave

  int id[W_ - 1];
#pragma unroll
  for (int j = 0; j < W_ - 1; ++j) id[j] = edgeToChk[e * (W_ - 1) + j];

#pragma unroll
  for (int j = 0; j < W_ - 1; ++j) {
    const float* src = llr_e + (size_t)id[j] * B + b4;
    const unsigned lds_off = (unsigned)(uintptr_t)(&stage[ty][j][tx]);
    asm volatile("global_load_async_to_lds_b128 %0, %1, off"
                 :
                 : "v"(lds_off), "v"(src)
                 : "memory");
  }
  asm volatile("s_wait_asynccnt 0x0" ::: "memory");

  const float a = alpha[0];
  unsigned sg[4] = {0u, 0u, 0u, 0u};
  float mn[4] = {INFINITY, INFINITY, INFINITY, INFINITY};
#pragma unroll
  for (int j = 0; j < W_ - 1; ++j) {
    const float4 x = stage[ty][j][tx];
    const float xc[4] = {x.x, x.y, x.z, x.w};
#pragma unroll
    for (int c = 0; c < 4; ++c) {
      sg[c] ^= __float_as_uint(xc[c]) & 0x80000000u;
      mn[c] = fminf(mn[c], fabsf(xc[c]));
    }
  }
  float4 r;
  r.x = __uint_as_float(__float_as_uint(a * mn[0]) ^ sg[0]);
  r.y = __uint_as_float(__float_as_uint(a * mn[1]) ^ sg[1]);
  r.z = __uint_as_float(__float_as_uint(a * mn[2]) ^ sg[2]);
  r.w = __uint_as_float(__float_as_uint(a * mn[3]) ^ sg[3]);
  *(float4*)(e2o + (size_t)e * B + b4) = r;
}

// Fused variable-node + output phase:
//   w2 = channel[v] + sum_w mask[v][w] * e2o[edge(v,w)]      (== next llr)
//   out[it][b][v] = sigmoid(-w2)                              (LDS transpose)
//   llr_e[edge(v,w)] = w2 - e2o[edge(v,w)]                    (next iteration)
__global__ void k_varout(const float* __restrict__ channelT,
                         const float* __restrict__ e2o,
                         const int* __restrict__ edgeToVar,
                         const float* __restrict__ mask,
                         float* __restrict__ out,
                         float* __restrict__ llr_e, int B, int write_llre) {
  __shared__ float tile[32][33];
  const int tx = threadIdx.x, ty = threadIdx.y;
  const int b = blockIdx.y * 32 + tx;
  const int v0 = blockIdx.x * 32;

  // Get all 24 gather rows in flight early (global_prefetch_b8: no LOADcnt).
#pragma unroll
  for (int k = 0; k < 4; ++k) {
    const int v = v0 + ty + 8 * k;
#pragma unroll
    for (int w = 0; w < W_; ++w)
      __builtin_prefetch(e2o + (size_t)edgeToVar[v * W_ + w] * B + b, 0, 3);
  }

#pragma unroll
  for (int k = 0; k < 4; ++k) {
    const int vl = ty + 8 * k;
    const int v = v0 + vl;
    float m[W_];
    int id[W_];
    float s = channelT[(size_t)v * B + b];
#pragma unroll
    for (int w = 0; w < W_; ++w) {
      id[w] = edgeToVar[v * W_ + w];
      m[w] = e2o[(size_t)id[w] * B + b];
      s = fmaf(m[w], mask[v * W_ + w], s);
    }
    if (write_llre) {
#pragma unroll
      for (int w = 0; w < W_; ++w)
        llr_e[(size_t)id[w] * B + b] = s - m[w];
    }
    tile[vl][tx] = s;
  }
  __syncthreads();
  // Transposed store: lanes now run along v -> contiguous 128B output stores.
#pragma unroll
  for (int k = 0; k < 4; ++k) {
    const int bb = blockIdx.y * 32 + ty + 8 * k;
    const int vv = v0 + tx;
    const float w2 = tile[tx][ty + 8 * k];
    out[(size_t)bb * N_ + vv] = 1.0f / (1.0f + __expf(w2));  // sigmoid(-w2)
  }
}

extern "C" void kernel_launch(void* const* d_in, const int* in_sizes, int n_in,
                              void* d_out, int out_size, void* d_ws,
                              size_t ws_size, hipStream_t stream) {
  (void)n_in;
  (void)ws_size;
  const float* channel   = (const float*)d_in[0];
  const int*   edgeToVar = (const int*)d_in[1];
  const float* mask      = (const float*)d_in[2];
  const int*   oddToEven = (const int*)d_in[3];
  const int*   edgeToChk = (const int*)d_in[4];
  const float* alpha     = (const float*)d_in[5];
  // d_in[6] is n_iters on device; derive it host-side from out_size instead
  // (reading device memory would require a sync and break graph capture).

  const int B = in_sizes[0] / N_;
  int iters = out_size / (B * N_);
  if (iters < 1) iters = 1;

  float* ws       = (float*)d_ws;
  float* channelT = ws;                          // N * B
  float* e2o      = channelT + (size_t)N_ * B;   // E * B
  float* llr_e    = e2o + (size_t)E_ * B;        // E * B
  float* outF     = (float*)d_out;

  const dim3 blk(32, 8);

  k_transpose<<<dim3(N_ / 32, B / 32), blk, 0, stream>>>(channel, channelT, B);
  k_init<<<dim3(E_ / 8, B / 128), blk, 0, stream>>>(channelT, oddToEven, llr_e,
                                                    B);

  for (int it = 0; it < iters; ++it) {
    k_chk<<<dim3(E_ / 8, B / 128), blk, 0, stream>>>(llr_e, edgeToChk, alpha,
                                                     e2o, B);
    k_varout<<<dim3(N_ / 32, B / 32), blk, 0, stream>>>(
        channelT, e2o, edgeToVar, mask, outF + (size_t)it * B * N_, llr_e, B,
        (it + 1 < iters) ? 1 : 0);
  }
}